// KEP_SVGPAttention_21921513079127
// MI455X (gfx1250) — compile-verified
//
#include <hip/hip_runtime.h>

// Problem constants (match reference)
#define BDIM   32
#define NSEQ   1024
#define CDIM   768
#define HHEADS 12
#define RLOW   10
#define MSUB   100
#define DH     64

// Big GEMM: qk[32768,1536] = X[32768,768] @ Wqk[1536,768]^T
#define GM (BDIM*NSEQ)   // 32768
#define GN (2*CDIM)      // 1536
#define GK (CDIM)        // 768
#define TM 128
#define TN 128
#define TK 32
#define LDA 36           // TK + 4 pad
#define LDB 132          // TN + 4 pad

typedef __attribute__((ext_vector_type(2))) float v2f;
typedef __attribute__((ext_vector_type(8))) float v8f;

// ---------------------------------------------------------------------------
// Kernel 1: qk = x @ Wqk^T via V_WMMA_F32_16X16X4_F32
// grid (GN/TN=12, GM/TM=256), block 256 (8 waves).
// Wave (mw = w&3, nw = w>>2) owns a 32x64 output footprint:
//   2 M-tiles (rows mw*32 .. +32) x 4 N-tiles (cols nw*64 .. +64)
// -> per K-step: 2 A-frags + 4 B-frag pairs feed 8 WMMAs (high LDS reuse).
// ---------------------------------------------------------------------------
__global__ __launch_bounds__(256)
void qk_gemm_wmma(const float* __restrict__ X, const float* __restrict__ W,
                  float* __restrict__ out) {
  __shared__ float As[TM * LDA];   // 18,432 B
  __shared__ float Bs[TK * LDB];   // 16,896 B
  const int tid  = threadIdx.x;
  const int lane = tid & 31;
  const int wave = tid >> 5;       // 0..7
  const int mw   = wave & 3;       // M group (32 rows each)
  const int nw   = wave >> 2;      // N group (64 cols each)
  const int lo   = lane & 15;
  const int hi   = lane >> 4;
  const int bm   = blockIdx.y * TM;
  const int bn   = blockIdx.x * TN;

  v8f acc00 = {0.f,0.f,0.f,0.f,0.f,0.f,0.f,0.f};
  v8f acc01 = acc00, acc02 = acc00, acc03 = acc00;
  v8f acc10 = acc00, acc11 = acc00, acc12 = acc00, acc13 = acc00;

  const int mrow0 = mw * 32 + lo;        // first M-tile row for this lane
  const int mrow1 = mrow0 + 16;          // second M-tile row
  const int cb    = nw * 64 + lo;        // column base within Bs

  for (int kk = 0; kk < GK; kk += TK) {
    // A tile: 128 rows x 32 cols (row-major, padded) -- 1024 float4
#pragma unroll
    for (int it = 0; it < 4; ++it) {
      int i   = tid + it * 256;
      int row = i >> 3;
      int c4  = (i & 7) << 2;
      float4 v = *(const float4*)(X + (size_t)(bm + row) * GK + kk + c4);
      *(float4*)(&As[row * LDA + c4]) = v;
    }
    // B tile: Bs[k][col] = Wqk[bn+col][kk+k] -- 1024 float4, transposed store
#pragma unroll
    for (int it = 0; it < 4; ++it) {
      int i  = tid + it * 256;
      int jl = i >> 3;             // 0..127
      int k4 = (i & 7) << 2;       // 0..28
      float4 v = *(const float4*)(W + (size_t)(bn + jl) * GK + kk + k4);
      Bs[(k4 + 0) * LDB + jl] = v.x;
      Bs[(k4 + 1) * LDB + jl] = v.y;
      Bs[(k4 + 2) * LDB + jl] = v.z;
      Bs[(k4 + 3) * LDB + jl] = v.w;
    }
    __syncthreads();

#pragma unroll
    for (int k4 = 0; k4 < 8; ++k4) {
      const int kb = k4 * 4 + hi * 2;  // ISA A layout: lanes 0-15 K={0,1}, 16-31 K={2,3}
      v2f a0, a1;
      a0.x = As[mrow0 * LDA + kb];  a0.y = As[mrow0 * LDA + kb + 1];
      a1.x = As[mrow1 * LDA + kb];  a1.y = As[mrow1 * LDA + kb + 1];
      v2f b0, b1, b2, b3;
      b0.x = Bs[kb * LDB + cb];        b0.y = Bs[(kb + 1) * LDB + cb];
      b1.x = Bs[kb * LDB + cb + 16];   b1.y = Bs[(kb + 1) * LDB + cb + 16];
      b2.x = Bs[kb * LDB + cb + 32];   b2.y = Bs[(kb + 1) * LDB + cb + 32];
      b3.x = Bs[kb * LDB + cb + 48];   b3.y = Bs[(kb + 1) * LDB + cb + 48];
      acc00 = __builtin_amdgcn_wmma_f32_16x16x4_f32(false, a0, false, b0, (short)0, acc00, false, false);
      acc01 = __builtin_amdgcn_wmma_f32_16x16x4_f32(false, a0, false, b1, (short)0, acc01, false, false);
      acc02 = __builtin_amdgcn_wmma_f32_16x16x4_f32(false, a0, false, b2, (short)0, acc02, false, false);
      acc03 = __builtin_amdgcn_wmma_f32_16x16x4_f32(false, a0, false, b3, (short)0, acc03, false, false);
      acc10 = __builtin_amdgcn_wmma_f32_16x16x4_f32(false, a1, false, b0, (short)0, acc10, false, false);
      acc11 = __builtin_amdgcn_wmma_f32_16x16x4_f32(false, a1, false, b1, (short)0, acc11, false, false);
      acc12 = __builtin_amdgcn_wmma_f32_16x16x4_f32(false, a1, false, b2, (short)0, acc12, false, false);
      acc13 = __builtin_amdgcn_wmma_f32_16x16x4_f32(false, a1, false, b3, (short)0, acc13, false, false);
    }
    __syncthreads();
  }

  // Store: C/D layout -> VGPR i holds M = i + 8*hi, N = lo
  const size_t r0 = (size_t)(bm + mw * 32 + 8 * hi);
  const int    cg = bn + nw * 64 + lo;
#pragma unroll
  for (int i = 0; i < 8; ++i) {
    float* p0 = out + (r0 + i) * GN + cg;
    p0[0]  = acc00[i];
    p0[16] = acc01[i];
    p0[32] = acc02[i];
    p0[48] = acc03[i];
    float* p1 = out + (r0 + 16 + i) * GN + cg;
    p1[0]  = acc10[i];
    p1[16] = acc11[i];
    p1[32] = acc12[i];
    p1[48] = acc13[i];
  }
}

// ---------------------------------------------------------------------------
// Kernel 2: we/wr = einsum('bhdm,hmr->bhdr', xs, we_p/wr_p)
// xs[b,h,d,m] = x[b, idx[m], h*64+d], idx = linspace(0,1023,100).astype(int)
// thread per (b,h,d,r); fused we & wr (shared xs loads).
// ---------------------------------------------------------------------------
__global__ __launch_bounds__(256)
void gen_weights(const float* __restrict__ x, const float* __restrict__ we_p,
                 const float* __restrict__ wr_p,
                 float* __restrict__ we, float* __restrict__ wr) {
  __shared__ int sidx[MSUB];
  const int tid = threadIdx.x;
  if (tid < MSUB) {
    const double delta = 1023.0 / 99.0;       // numpy linspace step (double)
    int v = (int)((double)tid * delta);       // truncation matches .astype(int32)
    if (tid == MSUB - 1) v = NSEQ - 1;        // numpy sets endpoint exactly
    sidx[tid] = v;
  }
  __syncthreads();

  const int g = blockIdx.x * 256 + tid;
  if (g >= BDIM * HHEADS * DH * RLOW) return;
  const int r = g % RLOW;
  const int d = (g / RLOW) % DH;
  const int h = (g / (RLOW * DH)) % HHEADS;
  const int b = g / (RLOW * DH * HHEADS);

  float aw = 0.f, ar = 0.f;
#pragma unroll 4
  for (int m = 0; m < MSUB; ++m) {
    float xs = x[((size_t)b * NSEQ + sidx[m]) * CDIM + h * DH + d];
    aw += xs * we_p[(h * MSUB + m) * RLOW + r];
    ar += xs * wr_p[(h * MSUB + m) * RLOW + r];
  }
  we[g] = aw;
  wr[g] = ar;
}

// ---------------------------------------------------------------------------
// Kernel 3: fused epilogue. One wave per (b,h,n):
//   normalize q/k -> escore/rscore -> v1 -> mean+noise -> samples -> attn
// block = 8 waves (8 seq positions), grid (NSEQ/8, H, B).
// ---------------------------------------------------------------------------
__device__ __forceinline__ float wave_sum32(float v) {
  v += __shfl_xor(v, 16, 32);
  v += __shfl_xor(v, 8, 32);
  v += __shfl_xor(v, 4, 32);
  v += __shfl_xor(v, 2, 32);
  v += __shfl_xor(v, 1, 32);
  return v;
}

__global__ __launch_bounds__(256)
void svgp_epilogue(const float* __restrict__ qk, const float* __restrict__ we,
                   const float* __restrict__ wr,
                   const float* __restrict__ log_lambda,
                   const float* __restrict__ m_u, const float* __restrict__ s_tril,
                   const float* __restrict__ log_ssqrt,
                   const float* __restrict__ Wf, const float* __restrict__ bf,
                   const float* __restrict__ eps,
                   float* __restrict__ esc, float* __restrict__ rsc,
                   float* __restrict__ attn) {
  __shared__ float we_s[DH * RLOW];
  __shared__ float wr_s[DH * RLOW];
  __shared__ float wf_s[DH * RLOW];
  __shared__ float sl_s[RLOW * RLOW * RLOW];
  __shared__ float mu_s[RLOW * RLOW];
  __shared__ float bf_s[DH];
  __shared__ float lam2_s[RLOW];

  const int tid  = threadIdx.x;
  const int lane = tid & 31;
  const int wave = tid >> 5;
  const int h = blockIdx.y;
  const int b = blockIdx.z;

  const size_t bh = (size_t)b * HHEADS + h;
  for (int i = tid; i < DH * RLOW; i += 256) {
    we_s[i] = we[bh * DH * RLOW + i];
    wr_s[i] = wr[bh * DH * RLOW + i];
    wf_s[i] = Wf[i];                               // [Dh,R] row-major
  }
  for (int i = tid; i < RLOW * RLOW * RLOW; i += 256) {
    int a = i / (RLOW * RLOW);
    int q = (i / RLOW) % RLOW;
    int c = i % RLOW;
    float v = (q == c) ? expf(log_ssqrt[(h * RLOW + a) * RLOW + q]) : 0.f;
    if (q > c) v += s_tril[((h * RLOW + a) * RLOW + q) * RLOW + c];
    sl_s[i] = v;                                   // sl[h][a][q][c]
  }
  for (int i = tid; i < RLOW * RLOW; i += 256) mu_s[i] = m_u[h * RLOW * RLOW + i];
  if (tid < DH)   bf_s[tid]   = bf[tid];
  if (tid < RLOW) lam2_s[tid] = expf(2.f * log_lambda[h * RLOW + tid]);
  __syncthreads();

  const int n = blockIdx.x * 8 + wave;
  const float* qrow = qk + ((size_t)b * NSEQ + n) * GN + h * DH;
  const float* krow = qrow + CDIM;

  float q0 = qrow[lane], q1 = qrow[lane + 32];
  float k0 = krow[lane], k1 = krow[lane + 32];
  float qn = wave_sum32(q0 * q0 + q1 * q1);
  float kn = wave_sum32(k0 * k0 + k1 * k1);
  float qi = 1.f / fmaxf(sqrtf(qn), 1e-12f);
  float ki = 1.f / fmaxf(sqrtf(kn), 1e-12f);
  q0 *= qi; q1 *= qi; k0 *= ki; k1 *= ki;

  float e[RLOW], rv[RLOW], v1[RLOW];
#pragma unroll
  for (int r = 0; r < RLOW; ++r) {
    e[r]  = wave_sum32(q0 * we_s[lane * RLOW + r] + q1 * we_s[(lane + 32) * RLOW + r]);
    rv[r] = wave_sum32(k0 * wr_s[lane * RLOW + r] + k1 * wr_s[(lane + 32) * RLOW + r]);
    v1[r] = (e[r] + rv[r]) * lam2_s[r];
  }

  const size_t so = (bh * NSEQ + n) * RLOW;
  if (lane == 0) {
#pragma unroll
    for (int r = 0; r < RLOW; ++r) { esc[so + r] = e[r]; rsc[so + r] = rv[r]; }
  }

  // lane c (< R) owns sample channel c
  float sample = 0.f;
  if (lane < RLOW) {
    const int c = lane;
    float mean = 0.f;
#pragma unroll
    for (int r = 0; r < RLOW; ++r) mean += v1[r] * mu_s[r * RLOW + c];
    const float* ep = eps + so;
    float noise = 0.f;
#pragma unroll
    for (int q = 0; q < RLOW; ++q) {
      float tq = 0.f;
#pragma unroll
      for (int a = 0; a < RLOW; ++a) tq += v1[a] * sl_s[(a * RLOW + q) * RLOW + c];
      noise += ep[q] * tq;
    }
    sample = mean + noise;
  }
  float sr[RLOW];
#pragma unroll
  for (int r = 0; r < RLOW; ++r) sr[r] = __shfl(sample, r, 32);

  float a0 = bf_s[lane], a1 = bf_s[lane + 32];
#pragma unroll
  for (int r = 0; r < RLOW; ++r) {
    a0 += sr[r] * wf_s[lane * RLOW + r];
    a1 += sr[r] * wf_s[(lane + 32) * RLOW + r];
  }
  float* ap = attn + ((size_t)b * NSEQ + n) * CDIM + h * DH;
  ap[lane]      = a0;
  ap[lane + 32] = a1;
}

// ---------------------------------------------------------------------------
// Kernel 4: KL scalar reduction (single block)
// ---------------------------------------------------------------------------
__global__ __launch_bounds__(256)
void kl_reduce(const float* __restrict__ log_lambda, const float* __restrict__ m_u,
               const float* __restrict__ s_tril, const float* __restrict__ log_ssqrt,
               float* __restrict__ out_kl) {
  __shared__ float red[256];
  const int tid = threadIdx.x;
  float p = 0.f;
  for (int i = tid; i < HHEADS * RLOW * RLOW * RLOW; i += 256) {
    int h = i / (RLOW * RLOW * RLOW);
    int a = (i / (RLOW * RLOW)) % RLOW;
    int q = (i / RLOW) % RLOW;
    int c = i % RLOW;
    float sl = (q == c) ? expf(log_ssqrt[(h * RLOW + a) * RLOW + q]) : 0.f;
    if (q > c) sl += s_tril[((h * RLOW + a) * RLOW + q) * RLOW + c];
    float t = expf(2.f * log_lambda[h * RLOW + a]) * sl;
    p += 0.5f * t * t;
  }
  for (int i = tid; i < HHEADS * RLOW * RLOW; i += 256) {
    int h = i / (RLOW * RLOW);
    int r = (i / RLOW) % RLOW;
    float mu = m_u[i];
    p += 0.5f * mu * mu * expf(4.f * log_lambda[h * RLOW + r]);
    p -= log_ssqrt[i];
  }
  for (int i = tid; i < HHEADS * RLOW; i += 256) p -= 2.f * (float)RLOW * log_lambda[i];
  red[tid] = p;
  __syncthreads();
  for (int s = 128; s > 0; s >>= 1) {
    if (tid < s) red[tid] += red[tid + s];
    __syncthreads();
  }
  if (tid == 0) out_kl[0] = red[0] - 0.5f * (float)(RLOW * RLOW * HHEADS);
}

// ---------------------------------------------------------------------------
extern "C" void kernel_launch(void* const* d_in, const int* in_sizes, int n_in,
                              void* d_out, int out_size, void* d_ws, size_t ws_size,
                              hipStream_t stream) {
  const float* x          = (const float*)d_in[0];
  const float* Wqk        = (const float*)d_in[1];
  const float* we_p       = (const float*)d_in[2];
  const float* wr_p       = (const float*)d_in[3];
  const float* log_lambda = (const float*)d_in[4];
  const float* m_u        = (const float*)d_in[5];
  const float* s_tril     = (const float*)d_in[6];
  const float* log_ssqrt  = (const float*)d_in[7];
  const float* Wf         = (const float*)d_in[8];
  const float* bf         = (const float*)d_in[9];
  const float* eps        = (const float*)d_in[10];

  // workspace layout (floats)
  float* ws = (float*)d_ws;
  float* qk = ws;                                  // 32768*1536 = 50,331,648
  float* we = ws + (size_t)GM * GN;                // 245,760
  float* wr = we + (size_t)BDIM * HHEADS * DH * RLOW;

  // output layout (tuple concat): attn | escore | rscore | kl
  float* attn = (float*)d_out;
  float* esc  = attn + (size_t)BDIM * NSEQ * CDIM;           // 25,165,824
  float* rsc  = esc + (size_t)BDIM * HHEADS * NSEQ * RLOW;   // +3,932,160
  float* klp  = rsc + (size_t)BDIM * HHEADS * NSEQ * RLOW;   // +3,932,160

  qk_gemm_wmma<<<dim3(GN / TN, GM / TM), 256, 0, stream>>>(x, Wqk, qk);
  gen_weights<<<(BDIM * HHEADS * DH * RLOW + 255) / 256, 256, 0, stream>>>(
      x, we_p, wr_p, we, wr);
  svgp_epilogue<<<dim3(NSEQ / 8, HHEADS, BDIM), 256, 0, stream>>>(
      qk, we, wr, log_lambda, m_u, s_tril, log_ssqrt, Wf, bf, eps, esc, rsc, attn);
  kl_reduce<<<1, 256, 0, stream>>>(log_lambda, m_u, s_tril, log_ssqrt, klp);
}